// GAT_1314259993086
// MI455X (gfx1250) — compile-verified
//
#include <hip/hip_runtime.h>
#include <hip/hip_bf16.h>

typedef __attribute__((ext_vector_type(2))) float v2f;
typedef __attribute__((ext_vector_type(8))) float v8f;

#define LRELU(v) ((v) > 0.f ? (v) : 0.2f * (v))

__device__ __forceinline__ unsigned f2mono(float f) {
    unsigned u = __float_as_uint(f);
    return (u & 0x80000000u) ? ~u : (u | 0x80000000u);
}
__device__ __forceinline__ float mono2f(unsigned u) {
    return (u & 0x80000000u) ? __uint_as_float(u & 0x7FFFFFFFu) : __uint_as_float(~u);
}

// ---------------- init ----------------
__global__ void k_init(float* sum_bn, unsigned* mx1, float* den1, unsigned* mx2,
                       float* den2, float* out2, const float* __restrict__ g2_bias, int N) {
    int i = blockIdx.x * blockDim.x + threadIdx.x;
    if (i < 257) { sum_bn[i] = 0.f; return; }           // [0]=sum_ea, [1..128]=bn sum, [129..256]=bn sumsq
    i -= 257;
    if (i < N * 8) { mx1[i] = 0u; return; }
    i -= N * 8;
    if (i < N * 8) { den1[i] = 0.f; return; }
    i -= N * 8;
    if (i < N) { mx2[i] = 0u; return; }
    i -= N;
    if (i < N) { den2[i] = 0.f; return; }
    i -= N;
    if (i < N * 16) { out2[i] = g2_bias[i & 15]; return; }
}

__global__ void k_sum_ea(const float* __restrict__ ea, int E, float* sum_out) {
    __shared__ float red[256];
    float acc = 0.f;
    for (int i = blockIdx.x * 256 + threadIdx.x; i < E; i += gridDim.x * 256) acc += ea[i];
    red[threadIdx.x] = acc;
    __syncthreads();
    for (int s = 128; s > 0; s >>= 1) {
        if ((int)threadIdx.x < s) red[threadIdx.x] += red[threadIdx.x + s];
        __syncthreads();
    }
    if (threadIdx.x == 0) atomicAdd(sum_out, red[0]);
}

// ---------------- layer-1 node linears via WMMA f32 16x16x4 (K padded 2->4) ----------------
__global__ void k_node_lin1(const float* __restrict__ x,
                            const float* __restrict__ Wl, const float* __restrict__ bl,
                            const float* __restrict__ Wr, const float* __restrict__ br,
                            const float* __restrict__ Ws, const float* __restrict__ bs,
                            const float* __restrict__ g1_bias,
                            float* __restrict__ xl1, float* __restrict__ xr1,
                            float* __restrict__ out1, int N) {
    int tile = (blockIdx.x * blockDim.x + threadIdx.x) >> 5;
    int lane = threadIdx.x & 31;
    int tiles = (N + 15) >> 4;
    if (tile >= tiles) return;                 // wave-uniform: EXEC stays all-ones for WMMA
    bool lo = lane < 16;
    int col0 = lane & 15;
    int r = tile * 16 + col0;
    int rc = r < N ? r : N - 1;
    bool full = (tile * 16 + 16) <= N;         // wave-uniform: true for every tile when N%16==0
    v2f a;                                     // A: 16x4, lanes0-15 K=0,1; lanes16-31 K=2,3 (zero pad)
    a.x = lo ? x[rc * 2 + 0] : 0.f;
    a.y = lo ? x[rc * 2 + 1] : 0.f;
    int rbase = tile * 16 + (lo ? 0 : 8);
    for (int nt = 0; nt < 8; ++nt) {
        int col = nt * 16 + col0;
        v2f bL, bR, bS;                        // B: 4x16, row K per half-wave; K=2,3 zero
        bL.x = lo ? Wl[col] : 0.f;  bL.y = lo ? Wl[128 + col] : 0.f;
        bR.x = lo ? Wr[col] : 0.f;  bR.y = lo ? Wr[128 + col] : 0.f;
        bS.x = lo ? Ws[col] : 0.f;  bS.y = lo ? Ws[128 + col] : 0.f;
        float cl = bl[col], cr = br[col], cs = bs[col] + g1_bias[col];
        v8f CL, CR, CS;
        #pragma unroll
        for (int j = 0; j < 8; ++j) { CL[j] = cl; CR[j] = cr; CS[j] = cs; }
        CL = __builtin_amdgcn_wmma_f32_16x16x4_f32(false, a, false, bL, (short)0, CL, false, false);
        CR = __builtin_amdgcn_wmma_f32_16x16x4_f32(false, a, false, bR, (short)0, CR, false, false);
        CS = __builtin_amdgcn_wmma_f32_16x16x4_f32(false, a, false, bS, (short)0, CS, false, false);
        if (full) {                            // uniform branch: unconditional stores, no exec dance
            #pragma unroll
            for (int j = 0; j < 8; ++j) {
                size_t row = (size_t)(rbase + j);
                xl1[row * 128 + col]  = CL[j];
                xr1[row * 128 + col]  = CR[j];
                out1[row * 128 + col] = CS[j];
            }
        } else {
            #pragma unroll
            for (int j = 0; j < 8; ++j) {
                int row = rbase + j;
                if (row < N) {
                    xl1[(size_t)row * 128 + col]  = CL[j];
                    xr1[(size_t)row * 128 + col]  = CR[j];
                    out1[(size_t)row * 128 + col] = CS[j];
                }
            }
        }
    }
}

// ---------------- layer-1 edge passes (H=8, C=16), thread = (edge, head) ----------------
__global__ void k_edge1_logits(const int* __restrict__ ei, const float* __restrict__ ea,
                               const float* __restrict__ sum_ea,
                               const float* __restrict__ xl1, const float* __restrict__ xr1,
                               const float* __restrict__ We, const float* __restrict__ att,
                               float* __restrict__ logits, unsigned* __restrict__ mx,
                               int N, int E) {
    int t = blockIdx.x * blockDim.x + threadIdx.x;
    int E2 = E + N;
    if (t >= E2 * 8) return;
    int e = t >> 3, hh = t & 7;
    int s, d; float ev;
    if (e < E) { s = ei[e]; d = ei[E + e]; ev = ea[e]; }
    else       { s = d = e - E; ev = sum_ea[0] / (float)E; }
    const float4* xls = (const float4*)(xl1 + (size_t)s * 128 + hh * 16);  // 64B-aligned
    const float4* xrd = (const float4*)(xr1 + (size_t)d * 128 + hh * 16);
    const float4* wv  = (const float4*)(We + hh * 16);
    const float4* av  = (const float4*)(att + hh * 16);
    float acc = 0.f;
    #pragma unroll
    for (int q = 0; q < 4; ++q) {
        float4 xa = xls[q], xb = xrd[q], w = wv[q], at = av[q];
        float v0 = xa.x + xb.x + ev * w.x;
        float v1 = xa.y + xb.y + ev * w.y;
        float v2 = xa.z + xb.z + ev * w.z;
        float v3 = xa.w + xb.w + ev * w.w;
        acc += at.x * LRELU(v0) + at.y * LRELU(v1) + at.z * LRELU(v2) + at.w * LRELU(v3);
    }
    logits[t] = acc;
    atomicMax(&mx[d * 8 + hh], f2mono(acc));
}

__global__ void k_edge1_exp(const int* __restrict__ ei, float* __restrict__ logits,
                            const unsigned* __restrict__ mx, float* __restrict__ den,
                            int N, int E) {
    int t = blockIdx.x * blockDim.x + threadIdx.x;
    int E2 = E + N;
    if (t >= E2 * 8) return;
    int e = t >> 3, hh = t & 7;
    int d = (e < E) ? ei[E + e] : (e - E);
    float ex = __expf(logits[t] - mono2f(mx[d * 8 + hh]));
    logits[t] = ex;                             // in-place: now holds exp
    atomicAdd(&den[d * 8 + hh], ex);
}

__global__ void k_edge1_scatter(const int* __restrict__ ei, const float* __restrict__ logits,
                                const float* __restrict__ den, const float* __restrict__ xl1,
                                float* __restrict__ out1, int N, int E) {
    int t = blockIdx.x * blockDim.x + threadIdx.x;
    int E2 = E + N;
    if (t >= E2 * 8) return;
    int e = t >> 3, hh = t & 7;
    int s, d;
    if (e < E) { s = ei[e]; d = ei[E + e]; } else { s = d = e - E; }
    float alpha = logits[t] / den[d * 8 + hh];
    const float4* xls = (const float4*)(xl1 + (size_t)s * 128 + hh * 16);  // b128 gathers
    float* od = out1 + (size_t)d * 128 + hh * 16;
    #pragma unroll
    for (int q = 0; q < 4; ++q) {
        float4 xa = xls[q];
        atomicAdd(&od[q * 4 + 0], xa.x * alpha);
        atomicAdd(&od[q * 4 + 1], xa.y * alpha);
        atomicAdd(&od[q * 4 + 2], xa.z * alpha);
        atomicAdd(&od[q * 4 + 3], xa.w * alpha);
    }
}

// ---------------- BatchNorm + ELU ----------------
__global__ void k_bn_reduce(const float* __restrict__ h, float* sums, float* sumsq, int N) {
    int g = blockIdx.x * blockDim.x + threadIdx.x;
    int c = g & 127;
    int slice = g >> 7;
    int nsl = (gridDim.x * blockDim.x) >> 7;
    float s = 0.f, q = 0.f;
    for (int r = slice; r < N; r += nsl) {
        float v = h[(size_t)r * 128 + c];
        s += v; q += v * v;
    }
    atomicAdd(&sums[c], s);
    atomicAdd(&sumsq[c], q);
}

__global__ void k_bn_apply(float* __restrict__ h, const float* __restrict__ sums,
                           const float* __restrict__ sumsq, const float* __restrict__ gamma,
                           const float* __restrict__ beta, int N) {
    int i = blockIdx.x * blockDim.x + threadIdx.x;          // one float4 per thread
    if (i >= N * 32) return;
    int c0 = (i * 4) & 127;
    float inv = 1.f / (float)N;
    float4 hv = ((float4*)h)[i];
    float4 sv = *(const float4*)(sums + c0);
    float4 qv = *(const float4*)(sumsq + c0);
    float4 gv = *(const float4*)(gamma + c0);
    float4 bv = *(const float4*)(beta + c0);
    float m0 = sv.x * inv, m1 = sv.y * inv, m2 = sv.z * inv, m3 = sv.w * inv;
    float v0 = (hv.x - m0) * rsqrtf(qv.x * inv - m0 * m0 + 1e-5f) * gv.x + bv.x;
    float v1 = (hv.y - m1) * rsqrtf(qv.y * inv - m1 * m1 + 1e-5f) * gv.y + bv.y;
    float v2 = (hv.z - m2) * rsqrtf(qv.z * inv - m2 * m2 + 1e-5f) * gv.z + bv.z;
    float v3 = (hv.w - m3) * rsqrtf(qv.w * inv - m3 * m3 + 1e-5f) * gv.w + bv.w;
    hv.x = v0 > 0.f ? v0 : (__expf(v0) - 1.f);
    hv.y = v1 > 0.f ? v1 : (__expf(v1) - 1.f);
    hv.z = v2 > 0.f ? v2 : (__expf(v2) - 1.f);
    hv.w = v3 > 0.f ? v3 : (__expf(v3) - 1.f);
    ((float4*)h)[i] = hv;
}

// ---------------- layer-2 node linears via WMMA: M=N, K=128, N=16 ----------------
__global__ void k_node_lin2(const float* __restrict__ h,
                            const float* __restrict__ Wl, const float* __restrict__ bl,
                            const float* __restrict__ Wr, const float* __restrict__ br,
                            float* __restrict__ xl2, float* __restrict__ xr2, int N) {
    int tile = (blockIdx.x * blockDim.x + threadIdx.x) >> 5;
    int lane = threadIdx.x & 31;
    int tiles = (N + 15) >> 4;
    if (tile >= tiles) return;                 // wave-uniform
    bool lo = lane < 16;
    int col = lane & 15;
    int r = tile * 16 + col;
    int rc = r < N ? r : N - 1;
    bool full = (tile * 16 + 16) <= N;         // wave-uniform
    int koff = lo ? 0 : 2;
    v8f CL, CR;
    float cl = bl[col], cr = br[col];
    #pragma unroll
    for (int j = 0; j < 8; ++j) { CL[j] = cl; CR[j] = cr; }
    const float* hrow = h + (size_t)rc * 128;
    #pragma unroll 4
    for (int k = 0; k < 128; k += 4) {
        v2f a = *(const v2f*)(hrow + k + koff);          // 8B-aligned pair load
        v2f bL, bR;
        bL.x = Wl[(k + koff) * 16 + col];  bL.y = Wl[(k + koff + 1) * 16 + col];
        bR.x = Wr[(k + koff) * 16 + col];  bR.y = Wr[(k + koff + 1) * 16 + col];
        CL = __builtin_amdgcn_wmma_f32_16x16x4_f32(false, a, false, bL, (short)0, CL, false, false);
        CR = __builtin_amdgcn_wmma_f32_16x16x4_f32(false, a, false, bR, (short)0, CR, false, false);
    }
    int rbase = tile * 16 + (lo ? 0 : 8);
    if (full) {
        #pragma unroll
        for (int j = 0; j < 8; ++j) {
            size_t row = (size_t)(rbase + j);
            xl2[row * 16 + col] = CL[j];
            xr2[row * 16 + col] = CR[j];
        }
    } else {
        #pragma unroll
        for (int j = 0; j < 8; ++j) {
            int row = rbase + j;
            if (row < N) {
                xl2[(size_t)row * 16 + col] = CL[j];
                xr2[(size_t)row * 16 + col] = CR[j];
            }
        }
    }
}

// ---------------- layer-2 edge passes (H=1, C=16), thread = edge ----------------
__global__ void k_edge2_logits(const int* __restrict__ ei, const float* __restrict__ ea,
                               const float* __restrict__ sum_ea,
                               const float* __restrict__ xl2, const float* __restrict__ xr2,
                               const float* __restrict__ We, const float* __restrict__ att,
                               float* __restrict__ logits, unsigned* __restrict__ mx,
                               int N, int E) {
    int e = blockIdx.x * blockDim.x + threadIdx.x;
    int E2 = E + N;
    if (e >= E2) return;
    int s, d; float ev;
    if (e < E) { s = ei[e]; d = ei[E + e]; ev = ea[e]; }
    else       { s = d = e - E; ev = sum_ea[0] / (float)E; }
    const float4* xls = (const float4*)(xl2 + (size_t)s * 16);
    const float4* xrd = (const float4*)(xr2 + (size_t)d * 16);
    const float4* wv  = (const float4*)We;
    const float4* av  = (const float4*)att;
    float acc = 0.f;
    #pragma unroll
    for (int q = 0; q < 4; ++q) {
        float4 xa = xls[q], xb = xrd[q], w = wv[q], at = av[q];
        float v0 = xa.x + xb.x + ev * w.x;
        float v1 = xa.y + xb.y + ev * w.y;
        float v2 = xa.z + xb.z + ev * w.z;
        float v3 = xa.w + xb.w + ev * w.w;
        acc += at.x * LRELU(v0) + at.y * LRELU(v1) + at.z * LRELU(v2) + at.w * LRELU(v3);
    }
    logits[e] = acc;
    atomicMax(&mx[d], f2mono(acc));
}

__global__ void k_edge2_exp(const int* __restrict__ ei, float* __restrict__ logits,
                            const unsigned* __restrict__ mx, float* __restrict__ den,
                            int N, int E) {
    int e = blockIdx.x * blockDim.x + threadIdx.x;
    int E2 = E + N;
    if (e >= E2) return;
    int d = (e < E) ? ei[E + e] : (e - E);
    float ex = __expf(logits[e] - mono2f(mx[d]));
    logits[e] = ex;
    atomicAdd(&den[d], ex);
}

__global__ void k_edge2_scatter(const int* __restrict__ ei, const float* __restrict__ logits,
                                const float* __restrict__ den, const float* __restrict__ xl2,
                                float* __restrict__ out2, float* __restrict__ alpha_out,
                                int N, int E) {
    int e = blockIdx.x * blockDim.x + threadIdx.x;
    int E2 = E + N;
    if (e >= E2) return;
    int s, d;
    if (e < E) { s = ei[e]; d = ei[E + e]; } else { s = d = e - E; }
    float alpha = logits[e] / den[d];
    alpha_out[e] = alpha;                       // second reference output
    const float4* xls = (const float4*)(xl2 + (size_t)s * 16);
    float* od = out2 + (size_t)d * 16;
    #pragma unroll
    for (int q = 0; q < 4; ++q) {
        float4 xa = xls[q];
        atomicAdd(&od[q * 4 + 0], xa.x * alpha);
        atomicAdd(&od[q * 4 + 1], xa.y * alpha);
        atomicAdd(&od[q * 4 + 2], xa.z * alpha);
        atomicAdd(&od[q * 4 + 3], xa.w * alpha);
    }
}

// ---------------- final heads: ELU + (16x16)->(16x9 | 16x7) ----------------
__global__ void k_heads(const float* __restrict__ out2,
                        const float* __restrict__ cc1W, const float* __restrict__ cc1b,
                        const float* __restrict__ cc2W, const float* __restrict__ cc2b,
                        const float* __restrict__ ct1W, const float* __restrict__ ct1b,
                        const float* __restrict__ ct2W, const float* __restrict__ ct2b,
                        float* __restrict__ out, int N) {
    int n = blockIdx.x * blockDim.x + threadIdx.x;
    if (n >= N) return;
    float hb[16], t1[16], u1[16];
    const float4* hv4 = (const float4*)(out2 + (size_t)n * 16);
    #pragma unroll
    for (int q = 0; q < 4; ++q) {
        float4 v = hv4[q];
        float a0 = v.x, a1 = v.y, a2 = v.z, a3 = v.w;
        hb[q * 4 + 0] = a0 > 0.f ? a0 : (__expf(a0) - 1.f);
        hb[q * 4 + 1] = a1 > 0.f ? a1 : (__expf(a1) - 1.f);
        hb[q * 4 + 2] = a2 > 0.f ? a2 : (__expf(a2) - 1.f);
        hb[q * 4 + 3] = a3 > 0.f ? a3 : (__expf(a3) - 1.f);
    }
    #pragma unroll
    for (int j = 0; j < 16; ++j) {
        float a = ct1b[j], b = cc1b[j];
        #pragma unroll
        for (int c = 0; c < 16; ++c) {
            a += hb[c] * ct1W[c * 16 + j];
            b += hb[c] * cc1W[c * 16 + j];
        }
        t1[j] = a; u1[j] = b;
    }
    float ob[16];
    #pragma unroll
    for (int j = 0; j < 9; ++j) {
        float a = cc2b[j];
        #pragma unroll
        for (int c = 0; c < 16; ++c) a += u1[c] * cc2W[c * 9 + j];
        ob[j] = a;                              // clone first
    }
    #pragma unroll
    for (int j = 0; j < 7; ++j) {
        float a = ct2b[j];
        #pragma unroll
        for (int c = 0; c < 16; ++c) a += t1[c] * ct2W[c * 7 + j];
        ob[9 + j] = a;                          // then type
    }
    float4* o4 = (float4*)(out + (size_t)n * 16);
    #pragma unroll
    for (int q = 0; q < 4; ++q)
        o4[q] = make_float4(ob[q * 4], ob[q * 4 + 1], ob[q * 4 + 2], ob[q * 4 + 3]);
}

extern "C" void kernel_launch(void* const* d_in, const int* in_sizes, int n_in,
                              void* d_out, int out_size, void* d_ws, size_t ws_size,
                              hipStream_t stream) {
    const float* x       = (const float*)d_in[0];
    const float* ea      = (const float*)d_in[1];
    const int*   ei      = (const int*)  d_in[2];
    const float* g1_Wl   = (const float*)d_in[3];
    const float* g1_bl   = (const float*)d_in[4];
    const float* g1_Wr   = (const float*)d_in[5];
    const float* g1_br   = (const float*)d_in[6];
    const float* g1_We   = (const float*)d_in[7];
    const float* g1_att  = (const float*)d_in[8];
    const float* g1_bias = (const float*)d_in[9];
    const float* skip_W  = (const float*)d_in[10];
    const float* skip_b  = (const float*)d_in[11];
    const float* bn_g    = (const float*)d_in[12];
    const float* bn_b    = (const float*)d_in[13];
    const float* g2_Wl   = (const float*)d_in[14];
    const float* g2_bl   = (const float*)d_in[15];
    const float* g2_Wr   = (const float*)d_in[16];
    const float* g2_br   = (const float*)d_in[17];
    const float* g2_We   = (const float*)d_in[18];
    const float* g2_att  = (const float*)d_in[19];
    const float* g2_bias = (const float*)d_in[20];
    const float* cc1_W   = (const float*)d_in[21];
    const float* cc1_b   = (const float*)d_in[22];
    const float* ct1_W   = (const float*)d_in[23];
    const float* ct1_b   = (const float*)d_in[24];
    const float* cc2_W   = (const float*)d_in[25];
    const float* cc2_b   = (const float*)d_in[26];
    const float* ct2_W   = (const float*)d_in[27];
    const float* ct2_b   = (const float*)d_in[28];

    const int N  = in_sizes[0] / 2;
    const int E  = in_sizes[1];
    const int E2 = E + N;

    float* ws = (float*)d_ws;
    float* sum_bn   = ws;          // [0]=sum_ea, [1..128]=bn sums, [129..256]=bn sumsq
    float* sum_ea   = ws;
    float* bn_sums  = ws + 1;
    float* bn_sumsq = ws + 129;
    size_t o = 512;
    float* xl1      = ws + o;  o += (size_t)N * 128;
    float* xr1      = ws + o;  o += (size_t)N * 128;
    float* out1     = ws + o;  o += (size_t)N * 128;
    float* logits1  = ws + o;  o += (size_t)E2 * 8;
    unsigned* mx1   = (unsigned*)(ws + o); o += (size_t)N * 8;
    float* den1     = ws + o;  o += (size_t)N * 8;
    float* xl2      = ws + o;  o += (size_t)N * 16;
    float* xr2      = ws + o;  o += (size_t)N * 16;
    float* logits2  = ws + o;  o += (size_t)E2;
    unsigned* mx2   = (unsigned*)(ws + o); o += (size_t)N;
    float* den2     = ws + o;  o += (size_t)N;
    float* out2     = ws + o;  o += (size_t)N * 16;

    float* outm   = (float*)d_out;          // [N,16]
    float* alpha2 = outm + (size_t)N * 16;  // [E2]

    const int B = 256;
    int initTot = 257 + N * 8 + N * 8 + N + N + N * 16;
    k_init<<<(initTot + B - 1) / B, B, 0, stream>>>(sum_bn, mx1, den1, mx2, den2, out2, g2_bias, N);
    k_sum_ea<<<256, 256, 0, stream>>>(ea, E, sum_ea);

    int tiles = (N + 15) / 16;
    int linThreads = tiles * 32;
    k_node_lin1<<<(linThreads + B - 1) / B, B, 0, stream>>>(x, g1_Wl, g1_bl, g1_Wr, g1_br,
                                                           skip_W, skip_b, g1_bias,
                                                           xl1, xr1, out1, N);
    int T1 = E2 * 8;
    k_edge1_logits<<<(T1 + B - 1) / B, B, 0, stream>>>(ei, ea, sum_ea, xl1, xr1, g1_We, g1_att,
                                                       logits1, mx1, N, E);
    k_edge1_exp<<<(T1 + B - 1) / B, B, 0, stream>>>(ei, logits1, mx1, den1, N, E);
    k_edge1_scatter<<<(T1 + B - 1) / B, B, 0, stream>>>(ei, logits1, den1, xl1, out1, N, E);

    k_bn_reduce<<<256, 256, 0, stream>>>(out1, bn_sums, bn_sumsq, N);
    int NV4 = N * 32;
    k_bn_apply<<<(NV4 + B - 1) / B, B, 0, stream>>>(out1, bn_sums, bn_sumsq, bn_g, bn_b, N);

    k_node_lin2<<<(linThreads + B - 1) / B, B, 0, stream>>>(out1, g2_Wl, g2_bl, g2_Wr, g2_br,
                                                            xl2, xr2, N);
    int T2 = E2;
    k_edge2_logits<<<(T2 + B - 1) / B, B, 0, stream>>>(ei, ea, sum_ea, xl2, xr2, g2_We, g2_att,
                                                       logits2, mx2, N, E);
    k_edge2_exp<<<(T2 + B - 1) / B, B, 0, stream>>>(ei, logits2, mx2, den2, N, E);
    k_edge2_scatter<<<(T2 + B - 1) / B, B, 0, stream>>>(ei, logits2, den2, xl2, out2, alpha2, N, E);

    k_heads<<<(N + B - 1) / B, B, 0, stream>>>(out2, cc1_W, cc1_b, cc2_W, cc2_b,
                                               ct1_W, ct1_b, ct2_W, ct2_b, outm, N);
}